// DynamicVocab_83554293776954
// MI455X (gfx1250) — compile-verified
//
#include <hip/hip_runtime.h>

// Problem sizes (fixed by the reference)
#define NN 32768
#define KK 2048
#define DD 1024

typedef float v2f __attribute__((ext_vector_type(2)));
typedef float v8f __attribute__((ext_vector_type(8)));

// ---- monotone float <-> unsigned encoding (order-preserving) ----
__device__ __forceinline__ unsigned encf(float f) {
  unsigned u = __float_as_uint(f);
  return (u & 0x80000000u) ? ~u : (u | 0x80000000u);
}
__device__ __forceinline__ float decf(unsigned k) {
  unsigned u = (k & 0x80000000u) ? (k & 0x7fffffffu) : ~k;
  return __uint_as_float(u);
}

// ---------------------------------------------------------------------------
// Kernel: encode subspace min/max inputs into output buffers, zero accumulators
// ---------------------------------------------------------------------------
__global__ void k_init(const float* __restrict__ smin_in,
                       const float* __restrict__ smax_in,
                       unsigned* __restrict__ smin_enc,
                       unsigned* __restrict__ smax_enc,
                       float* __restrict__ dist_sum,
                       int* __restrict__ hits) {
  size_t i = (size_t)blockIdx.x * blockDim.x + threadIdx.x;
  if (i < (size_t)KK * DD) {
    smin_enc[i] = encf(smin_in[i]);
    smax_enc[i] = encf(smax_in[i]);
  }
  if (i < KK) {
    dist_sum[i] = 0.0f;
    hits[i] = 0;
  }
}

// ---------------------------------------------------------------------------
// Kernel: |w|^2 per codeword (one wave per row; wave32 shuffle reduction)
// ---------------------------------------------------------------------------
__global__ __launch_bounds__(256) void k_wnorm(const float* __restrict__ table,
                                               float* __restrict__ wnorm) {
  const int wave = threadIdx.x >> 5;
  const int lane = threadIdx.x & 31;
  const int k = blockIdx.x * 8 + wave;
  const float* row = table + (size_t)k * DD;
  float s = 0.0f;
  for (int d = lane * 4; d < DD; d += 32 * 4) {
    float4 v = *(const float4*)(row + d);
    s += v.x * v.x + v.y * v.y + v.z * v.z + v.w * v.w;
  }
  #pragma unroll
  for (int off = 16; off; off >>= 1) s += __shfl_xor(s, off, 32);
  if (lane == 0) wnorm[k] = s;
}

// ---------------------------------------------------------------------------
// Fused GEMM (fp32 WMMA 16x16x4) + per-row argmin over all K codewords.
// MT=2 row tiles per wave (32 rows), KT=4 column tiles per K-step (64 cols).
// A fragment (16x4 f32): lane m=l&15 holds x[m][d + 2*(l>>4) + v], v in {0,1}.
// B fragment (4x16 f32): lane n=l&15 holds w[n][d + 2*(l>>4) + v].
// C tile layout: VGPR i, lanes0-15 -> (M=i, N=lane); lanes16-31 -> (M=i+8).
// Argmin: lane-local running min over packed u64 keys (enc(v)<<32 | column)
// inside the K loop (branchless VALU, co-issues with WMMA); the 16-lane
// cross-shuffle reduction is hoisted to kernel end (min is associative, and
// lexicographic key-min reproduces argmin's lowest-index tie-break exactly).
// ---------------------------------------------------------------------------
__device__ __forceinline__ unsigned long long
shfl_xor_u64(unsigned long long k, int off) {
  unsigned hi = (unsigned)(k >> 32);
  unsigned lo = (unsigned)k;
  hi = __shfl_xor(hi, off, 32);
  lo = __shfl_xor(lo, off, 32);
  return ((unsigned long long)hi << 32) | lo;
}

__device__ __forceinline__ void fold_tile(const v8f acc, float wn, int col,
                                          unsigned long long rkey[8]) {
  #pragma unroll
  for (int i = 0; i < 8; ++i) {
    const float v = wn - 2.0f * acc[i];  // |w|^2 - 2 x.w (row-const |x|^2 later)
    const unsigned long long k =
        ((unsigned long long)encf(v) << 32) | (unsigned)col;
    rkey[i] = (k < rkey[i]) ? k : rkey[i];  // branchless u64 min
  }
}

__global__ __launch_bounds__(128) void k_search(const float* __restrict__ x,
                                                const float* __restrict__ table,
                                                const float* __restrict__ wnorm,
                                                int* __restrict__ tokens,
                                                float* __restrict__ min_s) {
  const int lane = threadIdx.x & 31;
  const int wave = threadIdx.x >> 5;
  const int half = lane >> 4;
  const int nl = lane & 15;
  const int rb = blockIdx.x * 128 + wave * 32;  // this wave's 32-row tile base

  unsigned long long rkey0[8], rkey1[8];
  #pragma unroll
  for (int i = 0; i < 8; ++i) {
    rkey0[i] = ~0ull;
    rkey1[i] = ~0ull;
  }

  const float* aptr0 = x + (size_t)(rb + nl) * DD + 2 * half;
  const float* aptr1 = aptr0 + (size_t)16 * DD;

  for (int cb = 0; cb < KK; cb += 64) {
    v8f acc00 = {}, acc01 = {}, acc02 = {}, acc03 = {};
    v8f acc10 = {}, acc11 = {}, acc12 = {}, acc13 = {};
    const float* b0 = table + (size_t)(cb + nl) * DD + 2 * half;
    const float* b1 = b0 + 16 * DD;
    const float* b2 = b1 + 16 * DD;
    const float* b3 = b2 + 16 * DD;
    #pragma unroll 4
    for (int d = 0; d < DD; d += 4) {
      v2f a0  = *(const v2f*)(aptr0 + d);
      v2f a1  = *(const v2f*)(aptr1 + d);
      v2f vb0 = *(const v2f*)(b0 + d);
      v2f vb1 = *(const v2f*)(b1 + d);
      v2f vb2 = *(const v2f*)(b2 + d);
      v2f vb3 = *(const v2f*)(b3 + d);
      acc00 = __builtin_amdgcn_wmma_f32_16x16x4_f32(false, a0, false, vb0, (short)0, acc00, false, false);
      acc01 = __builtin_amdgcn_wmma_f32_16x16x4_f32(false, a0, false, vb1, (short)0, acc01, false, false);
      acc02 = __builtin_amdgcn_wmma_f32_16x16x4_f32(false, a0, false, vb2, (short)0, acc02, false, false);
      acc03 = __builtin_amdgcn_wmma_f32_16x16x4_f32(false, a0, false, vb3, (short)0, acc03, false, false);
      acc10 = __builtin_amdgcn_wmma_f32_16x16x4_f32(false, a1, false, vb0, (short)0, acc10, false, false);
      acc11 = __builtin_amdgcn_wmma_f32_16x16x4_f32(false, a1, false, vb1, (short)0, acc11, false, false);
      acc12 = __builtin_amdgcn_wmma_f32_16x16x4_f32(false, a1, false, vb2, (short)0, acc12, false, false);
      acc13 = __builtin_amdgcn_wmma_f32_16x16x4_f32(false, a1, false, vb3, (short)0, acc13, false, false);
    }
    const float wn0 = wnorm[cb +  0 + nl];
    const float wn1 = wnorm[cb + 16 + nl];
    const float wn2 = wnorm[cb + 32 + nl];
    const float wn3 = wnorm[cb + 48 + nl];
    fold_tile(acc00, wn0, cb +  0 + nl, rkey0);
    fold_tile(acc01, wn1, cb + 16 + nl, rkey0);
    fold_tile(acc02, wn2, cb + 32 + nl, rkey0);
    fold_tile(acc03, wn3, cb + 48 + nl, rkey0);
    fold_tile(acc10, wn0, cb +  0 + nl, rkey1);
    fold_tile(acc11, wn1, cb + 16 + nl, rkey1);
    fold_tile(acc12, wn2, cb + 32 + nl, rkey1);
    fold_tile(acc13, wn3, cb + 48 + nl, rkey1);
  }

  // single cross-lane (16-lane group) min reduction at the end
  #pragma unroll
  for (int i = 0; i < 8; ++i) {
    #pragma unroll
    for (int off = 1; off < 16; off <<= 1) {
      unsigned long long ok0 = shfl_xor_u64(rkey0[i], off);
      rkey0[i] = (ok0 < rkey0[i]) ? ok0 : rkey0[i];
      unsigned long long ok1 = shfl_xor_u64(rkey1[i], off);
      rkey1[i] = (ok1 < rkey1[i]) ? ok1 : rkey1[i];
    }
  }

  // lanes 0 and 16 each own 8 rows of each tile's final result
  if (nl == 0) {
    const int base0 = rb + 8 * half;
    const int base1 = rb + 16 + 8 * half;
    #pragma unroll
    for (int i = 0; i < 8; ++i) {
      tokens[base0 + i] = (int)(unsigned)rkey0[i];
      min_s[base0 + i] = decf((unsigned)(rkey0[i] >> 32));
      tokens[base1 + i] = (int)(unsigned)rkey1[i];
      min_s[base1 + i] = decf((unsigned)(rkey1[i] >> 32));
    }
  }
}

// ---------------------------------------------------------------------------
// Per-row segment update: |x|^2 reduction + atomic segment sum/min/max
// ---------------------------------------------------------------------------
__global__ __launch_bounds__(256) void k_update(const float* __restrict__ x,
                                                const int* __restrict__ tokens,
                                                const float* __restrict__ min_s,
                                                unsigned* __restrict__ smin_enc,
                                                unsigned* __restrict__ smax_enc,
                                                float* __restrict__ dist_sum,
                                                int* __restrict__ hits) {
  __shared__ float red[8];
  const int row = blockIdx.x;
  const int tid = threadIdx.x;
  const int t = tokens[row];
  const float* xr = x + (size_t)row * DD;
  unsigned* mn = smin_enc + (size_t)t * DD;
  unsigned* mx = smax_enc + (size_t)t * DD;

  const int d0 = tid * 4;
  float4 v = *(const float4*)(xr + d0);
  float s = v.x * v.x + v.y * v.y + v.z * v.z + v.w * v.w;

  atomicMin(&mn[d0 + 0], encf(v.x)); atomicMax(&mx[d0 + 0], encf(v.x));
  atomicMin(&mn[d0 + 1], encf(v.y)); atomicMax(&mx[d0 + 1], encf(v.y));
  atomicMin(&mn[d0 + 2], encf(v.z)); atomicMax(&mx[d0 + 2], encf(v.z));
  atomicMin(&mn[d0 + 3], encf(v.w)); atomicMax(&mx[d0 + 3], encf(v.w));

  #pragma unroll
  for (int off = 16; off; off >>= 1) s += __shfl_xor(s, off, 32);
  if ((tid & 31) == 0) red[tid >> 5] = s;
  __syncthreads();
  if (tid == 0) {
    float tot = 0.0f;
    #pragma unroll
    for (int i = 0; i < 8; ++i) tot += red[i];
    // min_dist = |x|^2 + (|w|^2 - 2 x.w)
    atomicAdd(&dist_sum[t], tot + min_s[row]);
    atomicAdd(&hits[t], 1);
  }
}

// ---------------------------------------------------------------------------
// Finalize: decode encoded min/max in place; compute distances/counts/unused
// ---------------------------------------------------------------------------
__global__ void k_final(float* __restrict__ out_min, float* __restrict__ out_max,
                        const float* __restrict__ dist_in,
                        const int* __restrict__ cnt_in,
                        const int* __restrict__ gu_in,
                        const float* __restrict__ dist_sum,
                        const int* __restrict__ hits,
                        float* __restrict__ out_dist,
                        int* __restrict__ out_cnt,
                        int* __restrict__ out_gu) {
  size_t i = (size_t)blockIdx.x * blockDim.x + threadIdx.x;
  if (i < (size_t)KK * DD) {
    unsigned kmn = ((const unsigned*)out_min)[i];
    unsigned kmx = ((const unsigned*)out_max)[i];
    out_min[i] = decf(kmn);
    out_max[i] = decf(kmx);
  }
  if (i < KK) {
    float cf = (float)cnt_in[i];
    float h = (float)hits[i];
    float denom = cf + h;
    out_dist[i] = (denom > 0.0f)
                      ? (cf * dist_in[i] + dist_sum[i]) / fmaxf(denom, 1.0f)
                      : dist_in[i];
    out_cnt[i] = cnt_in[i] + hits[i];
    out_gu[i] = (hits[i] > 0) ? 0 : gu_in[i];
  }
}

// ---------------------------------------------------------------------------
extern "C" void kernel_launch(void* const* d_in, const int* in_sizes, int n_in,
                              void* d_out, int out_size, void* d_ws, size_t ws_size,
                              hipStream_t stream) {
  const float* x        = (const float*)d_in[0];  // [N, D]
  const float* table    = (const float*)d_in[1];  // [K, D]
  const float* dist_in  = (const float*)d_in[2];  // [K]
  const int*   cnt_in   = (const int*)d_in[3];    // [K]
  const int*   gu_in    = (const int*)d_in[4];    // [K]
  const float* smin_in  = (const float*)d_in[5];  // [K, D]
  const float* smax_in  = (const float*)d_in[6];  // [K, D]

  // outputs, flat in return order
  float* out      = (float*)d_out;
  int*   o_tokens = (int*)out;                          // [N] int32
  float* o_dist   = out + NN;                           // [K] f32
  int*   o_cnt    = (int*)(out + NN + (size_t)KK);      // [K] int32
  int*   o_gu     = (int*)(out + NN + (size_t)2 * KK);  // [K] int32
  float* o_smin   = out + NN + (size_t)3 * KK;          // [K, D] f32
  float* o_smax   = o_smin + (size_t)KK * DD;           // [K, D] f32

  // workspace
  float* wnorm    = (float*)d_ws;           // [K]
  float* min_s    = wnorm + KK;             // [N]  (|w|^2 - 2 x.w at argmin)
  float* dist_sum = min_s + NN;             // [K]
  int*   hits     = (int*)(dist_sum + KK);  // [K]

  const size_t kd = (size_t)KK * DD;
  const int grid_kd = (int)((kd + 255) / 256);

  k_init<<<grid_kd, 256, 0, stream>>>(smin_in, smax_in,
                                      (unsigned*)o_smin, (unsigned*)o_smax,
                                      dist_sum, hits);
  k_wnorm<<<KK / 8, 256, 0, stream>>>(table, wnorm);
  k_search<<<NN / 128, 128, 0, stream>>>(x, table, wnorm, o_tokens, min_s);
  k_update<<<NN, 256, 0, stream>>>(x, o_tokens, min_s,
                                   (unsigned*)o_smin, (unsigned*)o_smax,
                                   dist_sum, hits);
  k_final<<<grid_kd, 256, 0, stream>>>(o_smin, o_smax, dist_in, cnt_in, gu_in,
                                       dist_sum, hits, o_dist, o_cnt, o_gu);
}